// spuLayerTransformer_64321430225053
// MI455X (gfx1250) — compile-verified
//
#include <hip/hip_runtime.h>

// MI455X (gfx1250) — memory-bound elementwise SPU bound propagation.
// 384 MB streamed @ 23.3 TB/s => ~16.5 us floor. Strategy:
//  * b128 (float4) non-temporal loads/stores: wave32 x 16B = 512B contiguous
//    per wave-access, full cachelines, th:NT since every line is touched once
//    and the 384MB working set exceeds the 192MB L2.
//  * Branchless v_cndmask selection of the three bound cases (no divergence).
//  * One v_exp_f32 + one v_rcp_f32 per spu() evaluation (4 evals/element).

typedef __attribute__((ext_vector_type(4))) float v4f;

__device__ __forceinline__ float spu1(float v) {
    // spu(v) = v*v - 0.5           for v >= 0
    //        = sigmoid(-v) - 1     for v <  0   (== 1/(1+e^v) - 1)
    float e   = __expf(v);                       // v_mul + v_exp_f32
    float sg  = __builtin_amdgcn_rcpf(1.0f + e); // v_rcp_f32
    float neg = sg - 1.0f;
    float pos = __fmaf_rn(v, v, -0.5f);
    return (v >= 0.0f) ? pos : neg;              // v_cmp + v_cndmask
}

__device__ __forceinline__ void spu_triplet(float x, float l, float u,
                                            float& ox, float& ol, float& ou) {
    float sx  = spu1(x);
    float sl  = spu1(l);
    float su  = spu1(u);
    float ssu = spu1(su);        // faithful to reference: u<=0 branch uses spu(spu(u))
    bool lge0 = (l >= 0.0f);
    bool ule0 = (u <= 0.0f);
    ox = sx;
    ol = lge0 ? sl : (ule0 ? su  : -0.5f);
    ou = lge0 ? su : (ule0 ? ssu : su);
}

__global__ __launch_bounds__(256) void spu_bounds_v4_kernel(
    const v4f* __restrict__ xp, const v4f* __restrict__ lp,
    const v4f* __restrict__ up,
    v4f* __restrict__ oxp, v4f* __restrict__ olp, v4f* __restrict__ oup,
    int n4)
{
    int i = blockIdx.x * blockDim.x + threadIdx.x;
    if (i >= n4) return;

    v4f xv = __builtin_nontemporal_load(xp + i);   // global_load_b128 th:NT
    v4f lv = __builtin_nontemporal_load(lp + i);
    v4f uv = __builtin_nontemporal_load(up + i);

    v4f ox, ol, ou;
#pragma unroll
    for (int c = 0; c < 4; ++c) {
        float a, b, d;
        spu_triplet(xv[c], lv[c], uv[c], a, b, d);
        ox[c] = a; ol[c] = b; ou[c] = d;
    }

    __builtin_nontemporal_store(ox, oxp + i);      // global_store_b128 th:NT
    __builtin_nontemporal_store(ol, olp + i);
    __builtin_nontemporal_store(ou, oup + i);
}

// Scalar tail for n % 4 != 0 (not hit for DIM = 16M, but keeps launch general).
__global__ void spu_bounds_tail_kernel(
    const float* __restrict__ x, const float* __restrict__ l,
    const float* __restrict__ u,
    float* __restrict__ ox, float* __restrict__ ol, float* __restrict__ ou,
    int start, int n)
{
    int i = start + blockIdx.x * blockDim.x + threadIdx.x;
    if (i >= n) return;
    float a, b, d;
    spu_triplet(x[i], l[i], u[i], a, b, d);
    ox[i] = a; ol[i] = b; ou[i] = d;
}

extern "C" void kernel_launch(void* const* d_in, const int* in_sizes, int n_in,
                              void* d_out, int out_size, void* d_ws, size_t ws_size,
                              hipStream_t stream)
{
    const float* x = (const float*)d_in[0];   // x
    const float* l = (const float*)d_in[1];   // l_in
    const float* u = (const float*)d_in[2];   // u_in
    float* out = (float*)d_out;               // [x_out | l_out | u_out] concat

    const int n  = in_sizes[0];
    const int n4 = n >> 2;

    float* ox = out;
    float* ol = out + (size_t)n;
    float* ou = out + 2 * (size_t)n;

    if (n4 > 0) {
        const int block = 256;                 // 8 wave32 waves per workgroup
        const int grid  = (n4 + block - 1) / block;
        spu_bounds_v4_kernel<<<grid, block, 0, stream>>>(
            (const v4f*)x, (const v4f*)l, (const v4f*)u,
            (v4f*)ox, (v4f*)ol, (v4f*)ou, n4);
    }

    const int start = n4 << 2;
    const int rem   = n - start;
    if (rem > 0) {
        spu_bounds_tail_kernel<<<(rem + 63) / 64, 64, 0, stream>>>(
            x, l, u, ox, ol, ou, start, n);
    }
}